// ChebNet_57621281243165
// MI455X (gfx1250) — compile-verified
//
#include <hip/hip_runtime.h>
#include <math.h>

// ---------------------------------------------------------------------------
// ChebNet (K=3) forward for MI455X / gfx1250.
//
// Cost model: 4 edge-scatter passes ~= 2 GB of HBM traffic (the bottleneck at
// 23.3 TB/s); GEMMs are only ~6.4 GFLOP total, so we run them in full FP32 on
// the matrix unit via V_WMMA_F32_16X16X4_F32 for reference-level numerics.
// ---------------------------------------------------------------------------

typedef __attribute__((ext_vector_type(2))) float v2f;
typedef __attribute__((ext_vector_type(8))) float v8f;

#define N0   65536
#define E0   1048576
#define N1   32768
#define E1   524288
#define FIN  64
#define FH   128
#define NOUT 10

// ---- degree / normalization ------------------------------------------------

__global__ void deg_kernel(const int* __restrict__ dst, float* __restrict__ deg, int E) {
  int t = blockIdx.x * blockDim.x + threadIdx.x;
  if (t < E) atomicAdd(&deg[dst[t]], 1.0f);
}

__global__ void dinv_kernel(float* __restrict__ deg, int N) {
  int t = blockIdx.x * blockDim.x + threadIdx.x;
  if (t < N) deg[t] = 1.0f / sqrtf(fmaxf(deg[t], 1.0f));
}

// ---- Xt assembly -----------------------------------------------------------

// Xt0[n, 0:64] = feat, Xt0[n, 64:192] = 0
__global__ void init_xt0_kernel(const float* __restrict__ feat, float* __restrict__ Xt) {
  long long t = blockIdx.x * (long long)blockDim.x + threadIdx.x;
  if (t >= (long long)N0 * 3 * FIN) return;
  int n = (int)(t / (3 * FIN));
  int f = (int)(t % (3 * FIN));
  Xt[t] = (f < FIN) ? feat[(long long)n * FIN + f] : 0.0f;
}

// zero columns [F, 3F) of an [N, 3F] buffer
__global__ void zero_cols_kernel(float* __restrict__ Xt, int N, int F) {
  long long t = blockIdx.x * (long long)blockDim.x + threadIdx.x;
  long long total = (long long)N * 2 * F;
  if (t >= total) return;
  int n = (int)(t / (2 * F));
  int j = (int)(t % (2 * F));
  Xt[(long long)n * 3 * F + F + j] = 0.0f;
}

// ---- edge propagation: out[dst] += x[src] * dinv[src]  (raw segment sum) ---

__global__ void scatter_kernel(const float* __restrict__ x, float* __restrict__ out,
                               const float* __restrict__ dinv,
                               const int* __restrict__ src, const int* __restrict__ dst,
                               int E, int F, int ld) {
  int F4 = F >> 2;
  long long t = blockIdx.x * (long long)blockDim.x + threadIdx.x;
  long long total = (long long)E * F4;
  if (t >= total) return;
  int e  = (int)(t / F4);
  int fc = ((int)(t % F4)) << 2;
  int s = src[e], d = dst[e];
  float ds = dinv[s];
  float4 v = *(const float4*)(x + (long long)s * ld + fc);
  float* o = out + (long long)d * ld + fc;
  atomicAdd(o + 0, v.x * ds);
  atomicAdd(o + 1, v.y * ds);
  atomicAdd(o + 2, v.z * ds);
  atomicAdd(o + 3, v.w * ds);
}

// X1 = -(acc * dinv[n])   (acc lives in cols [F, 2F))
__global__ void scale1_kernel(float* __restrict__ Xt, const float* __restrict__ dinv,
                              int N, int F) {
  long long t = blockIdx.x * (long long)blockDim.x + threadIdx.x;
  if (t >= (long long)N * F) return;
  int n = (int)(t / F);
  int f = (int)(t % F);
  float* p = Xt + (long long)n * 3 * F + F + f;
  *p = -(*p) * dinv[n];
}

// X2 = -2*acc*dinv[n] - X0   (acc lives in cols [2F, 3F))
__global__ void scale2_kernel(float* __restrict__ Xt, const float* __restrict__ dinv,
                              int N, int F) {
  long long t = blockIdx.x * (long long)blockDim.x + threadIdx.x;
  if (t >= (long long)N * F) return;
  int n = (int)(t / F);
  int f = (int)(t % F);
  long long row = (long long)n * 3 * F;
  float* p = Xt + row + 2 * F + f;
  *p = -2.0f * (*p) * dinv[n] - Xt[row + f];
}

// ---- WMMA GEMM: out[M,128] = relu(X[M,KDIM] @ W[KDIM,128] + bias) ----------
// One wave per 16x16 output tile; K stepped by 4 with V_WMMA_F32_16X16X4_F32.
// A 16x4 f32 layout: lanes 0-15 hold K=0,1; lanes 16-31 hold K=2,3 (M = lane&15).
// B 4x16 mirrored by N. C/D: VGPR i -> row i + 8*(lane>>4), col = lane&15.

template <int KDIM>
__global__ __launch_bounds__(256) void gemm_wmma_relu(const float* __restrict__ X,
                                                      const float* __restrict__ W,
                                                      const float* __restrict__ bias,
                                                      float* __restrict__ out, int M) {
  int wid  = (blockIdx.x * blockDim.x + threadIdx.x) >> 5;
  int lane = threadIdx.x & 31;
  int mtile = wid >> 3;        // 8 N-tiles cover the 128 output columns
  int ntile = wid & 7;
  int m0 = mtile << 4;
  int n0 = ntile << 4;
  if (m0 >= M) return;         // wave-uniform (grid sized exactly)
  int half = lane >> 4;
  int r    = lane & 15;

  const float* arow = X + (long long)(m0 + r) * KDIM + (half << 1);
  const float* bcol = W + ((half << 1) * FH) + n0 + r;

  v8f c = {};
#pragma unroll
  for (int k = 0; k < KDIM; k += 4) {
    v2f a, b;
    a.x = arow[k];
    a.y = arow[k + 1];
    b.x = bcol[k * FH];
    b.y = bcol[k * FH + FH];
    c = __builtin_amdgcn_wmma_f32_16x16x4_f32(false, a, false, b, (short)0, c,
                                              false, false);
  }

  float bb = bias[n0 + r];
#pragma unroll
  for (int i = 0; i < 8; ++i) {
    float v = c[i] + bb;
    out[(long long)(m0 + (half << 3) + i) * FH + n0 + r] = fmaxf(v, 0.0f);
  }
}

// ---- pooling / readout -----------------------------------------------------

// out[n, f] = max(in[2n, f], in[2n+1, f]); out has row stride ldo
__global__ void pool_kernel(const float* __restrict__ in, float* __restrict__ out,
                            int Nout, int ldo) {
  long long t = blockIdx.x * (long long)blockDim.x + threadIdx.x;
  if (t >= (long long)Nout * FH) return;
  int n = (int)(t / FH);
  int f = (int)(t % FH);
  float a = in[(long long)(2 * n) * FH + f];
  float b = in[(long long)(2 * n + 1) * FH + f];
  out[(long long)n * ldo + f] = fmaxf(a, b);
}

__global__ void gmax_kernel(const float* __restrict__ P, float* __restrict__ g, int N) {
  int f = threadIdx.x;  // 128 threads
  float m = -3.402823e38f;
  for (int i = 0; i < N; ++i) m = fmaxf(m, P[(long long)i * FH + f]);
  g[f] = m;
}

__global__ void head_kernel(const float* __restrict__ g, const float* __restrict__ Wc,
                            const float* __restrict__ bc, float* __restrict__ out) {
  __shared__ float l[NOUT];
  int t = threadIdx.x;  // 32 threads
  if (t < NOUT) {
    float s = bc[t];
    for (int h = 0; h < FH; ++h) s += g[h] * Wc[h * NOUT + t];
    l[t] = s;
  }
  __syncthreads();
  if (t == 0) {
    float m = l[0];
    for (int i = 1; i < NOUT; ++i) m = fmaxf(m, l[i]);
    float s = 0.0f;
    for (int i = 0; i < NOUT; ++i) s += expf(l[i] - m);
    float ls = logf(s);
    for (int i = 0; i < NOUT; ++i) out[i] = l[i] - m - ls;
  }
}

// ---------------------------------------------------------------------------

extern "C" void kernel_launch(void* const* d_in, const int* in_sizes, int n_in,
                              void* d_out, int out_size, void* d_ws, size_t ws_size,
                              hipStream_t stream) {
  const float* feat = (const float*)d_in[0];
  const int*   src0 = (const int*)d_in[1];
  const int*   dst0 = (const int*)d_in[2];
  const int*   src1 = (const int*)d_in[3];
  const int*   dst1 = (const int*)d_in[4];
  const float* W0   = (const float*)d_in[5];
  const float* b0   = (const float*)d_in[6];
  const float* W1   = (const float*)d_in[7];
  const float* b1   = (const float*)d_in[8];
  const float* Wc   = (const float*)d_in[9];
  const float* bc   = (const float*)d_in[10];

  // Workspace layout (floats); total ~= 80.4 MiB.
  // bufA: Xt0 [N0 x 192] then (aliased) Xt1 [N1 x 384] -- identical sizes.
  // bufB: H0 [N0 x 128]; later H1 [N1 x 128] at offset 0 and P1 at +N1*128.
  float* ws    = (float*)d_ws;
  float* dinv0 = ws;
  float* dinv1 = dinv0 + N0;
  float* bufA  = dinv1 + N1;
  float* bufB  = bufA + (size_t)N0 * 3 * FIN;
  float* Xt0 = bufA;
  float* Xt1 = bufA;
  float* H0  = bufB;
  float* H1  = bufB;
  float* P1  = bufB + (size_t)N1 * FH;          // inside old H0 region, after H1
  float* g   = bufB + (size_t)N0 * FH;

  const int TB = 256;
  // degrees -> D^{-1/2}
  hipMemsetAsync(dinv0, 0, (size_t)(N0 + N1) * sizeof(float), stream);
  deg_kernel<<<(E0 + TB - 1) / TB, TB, 0, stream>>>(dst0, dinv0, E0);
  deg_kernel<<<(E1 + TB - 1) / TB, TB, 0, stream>>>(dst1, dinv1, E1);
  dinv_kernel<<<(N0 + TB - 1) / TB, TB, 0, stream>>>(dinv0, N0);
  dinv_kernel<<<(N1 + TB - 1) / TB, TB, 0, stream>>>(dinv1, N1);

  // ---------------- layer 0 ----------------
  {
    long long tot = (long long)N0 * 3 * FIN;
    init_xt0_kernel<<<(int)((tot + TB - 1) / TB), TB, 0, stream>>>(feat, Xt0);
  }
  {
    long long tot = (long long)E0 * (FIN / 4);
    int blocks = (int)((tot + TB - 1) / TB);
    scatter_kernel<<<blocks, TB, 0, stream>>>(Xt0, Xt0 + FIN, dinv0, src0, dst0,
                                              E0, FIN, 3 * FIN);
    scale1_kernel<<<((long long)N0 * FIN + TB - 1) / TB, TB, 0, stream>>>(Xt0, dinv0, N0, FIN);
    scatter_kernel<<<blocks, TB, 0, stream>>>(Xt0 + FIN, Xt0 + 2 * FIN, dinv0, src0, dst0,
                                              E0, FIN, 3 * FIN);
    scale2_kernel<<<((long long)N0 * FIN + TB - 1) / TB, TB, 0, stream>>>(Xt0, dinv0, N0, FIN);
  }
  gemm_wmma_relu<3 * FIN><<<(N0 / 16) * 8 * 32 / TB, TB, 0, stream>>>(Xt0, W0, b0, H0, N0);

  // pool into Xt1's X0 slice; zero X1/X2 slices
  pool_kernel<<<((long long)N1 * FH + TB - 1) / TB, TB, 0, stream>>>(H0, Xt1, N1, 3 * FH);
  zero_cols_kernel<<<((long long)N1 * 2 * FH + TB - 1) / TB, TB, 0, stream>>>(Xt1, N1, FH);

  // ---------------- layer 1 ----------------
  {
    long long tot = (long long)E1 * (FH / 4);
    int blocks = (int)((tot + TB - 1) / TB);
    scatter_kernel<<<blocks, TB, 0, stream>>>(Xt1, Xt1 + FH, dinv1, src1, dst1,
                                              E1, FH, 3 * FH);
    scale1_kernel<<<((long long)N1 * FH + TB - 1) / TB, TB, 0, stream>>>(Xt1, dinv1, N1, FH);
    scatter_kernel<<<blocks, TB, 0, stream>>>(Xt1 + FH, Xt1 + 2 * FH, dinv1, src1, dst1,
                                              E1, FH, 3 * FH);
    scale2_kernel<<<((long long)N1 * FH + TB - 1) / TB, TB, 0, stream>>>(Xt1, dinv1, N1, FH);
  }
  gemm_wmma_relu<3 * FH><<<(N1 / 16) * 8 * 32 / TB, TB, 0, stream>>>(Xt1, W1, b1, H1, N1);

  // pool, global max readout, classifier head
  pool_kernel<<<((long long)(N1 / 2) * FH + TB - 1) / TB, TB, 0, stream>>>(H1, P1, N1 / 2, FH);
  gmax_kernel<<<1, FH, 0, stream>>>(P1, g, N1 / 2);
  head_kernel<<<1, 32, 0, stream>>>(g, Wc, bc, (float*)d_out);
}